// DNNTSP_NoTDL_80487687127273
// MI455X (gfx1250) — compile-verified
//
#include <hip/hip_runtime.h>

// GCN-conv + mean-pool, algebraically collapsed (see analysis):
//   out = (1/N) * (Σ_n coef[n] * emb[node_seq[n]]) @ W + b
// coef[n] = dis[n]*s[n] + 1/deg[n]
// deg[n]  = 1 + Σ_{e: col[e]==n} w[e]           (self-loop weight 1)
// s[n]    = Σ_{e: row[e]==n} w[e]*dis[col[e]]
// Memory-bound (~50 MB traffic @ 23.3 TB/s); final 1x64 @ 64x32 projection
// done with V_WMMA_F32_16X16X4_F32 in a single wave32 (fp32 end-to-end).

typedef __attribute__((ext_vector_type(2))) float v2f;
typedef __attribute__((ext_vector_type(8))) float v8f;

#define EMB_DIM 64
#define OUT_DIM 32

// ---------------- init: deg=1 (self loop), s=0, acc64=0 ----------------
__global__ void k_init(float* __restrict__ deg, float* __restrict__ s,
                       float* __restrict__ acc64, int N) {
    int i = blockIdx.x * blockDim.x + threadIdx.x;
    if (i < N) { deg[i] = 1.0f; s[i] = 0.0f; }
    if (i < EMB_DIM) acc64[i] = 0.0f;
}

// ---------------- deg[col[e]] += w[e] ----------------
__global__ void k_deg(const int* __restrict__ col, const float* __restrict__ w,
                      float* __restrict__ deg, int E) {
    int e = blockIdx.x * blockDim.x + threadIdx.x;
    if (e < E) atomicAdd(&deg[col[e]], w[e]);
}

// ---------------- dis = deg^{-1/2} ----------------
__global__ void k_dis(const float* __restrict__ deg, float* __restrict__ dis, int N) {
    int i = blockIdx.x * blockDim.x + threadIdx.x;
    if (i < N) {
        float d = deg[i];
        dis[i] = (d > 0.0f) ? rsqrtf(d) : 0.0f;
    }
}

// ---------------- s[row[e]] += w[e]*dis[col[e]] ----------------
__global__ void k_s(const int* __restrict__ row, const int* __restrict__ col,
                    const float* __restrict__ w, const float* __restrict__ dis,
                    float* __restrict__ s, int E) {
    int e = blockIdx.x * blockDim.x + threadIdx.x;
    if (e < E) atomicAdd(&s[row[e]], w[e] * dis[col[e]]);
}

// ---------------- acc64[d] += Σ_n coef[n] * emb[node_seq[n]][d] ----------------
// 256 threads = 4 groups of 64; each group streams nodes, thread t handles
// dim t%64 (coalesced 256B row reads of the L2-resident embedding table).
// Reduce group partials through LDS (ds_add_f32), then 64 global atomics/block.
__global__ void k_nodesum(const int* __restrict__ node_seq,
                          const float* __restrict__ emb,
                          const float* __restrict__ deg,
                          const float* __restrict__ dis,
                          const float* __restrict__ s,
                          float* __restrict__ acc64, int N) {
    __shared__ float sacc[EMB_DIM];
    const int t = threadIdx.x;
    const int d = t & (EMB_DIM - 1);
    const int g = t >> 6;                         // group 0..3
    if (t < EMB_DIM) sacc[t] = 0.0f;
    __syncthreads();

    float local = 0.0f;
    const int stride = gridDim.x * 4;
    for (int n = blockIdx.x * 4 + g; n < N; n += stride) {
        float coef = dis[n] * s[n] + 1.0f / deg[n];
        int tok = node_seq[n];
        local += coef * emb[tok * EMB_DIM + d];
    }
    atomicAdd(&sacc[d], local);                    // ds_add_f32
    __syncthreads();
    if (t < EMB_DIM) atomicAdd(&acc64[t], sacc[t]);
}

// ---------------- out = (acc64 @ W) * (1/N) + b, via V_WMMA_F32_16X16X4_F32 ----
// Single wave32, EXEC all ones. A(16x4) f32 layout (ISA 7.12.2): lanes 0-15
// hold M=lane, {VGPR0,VGPR1}={K0,K1}; lanes 16-31 hold {K2,K3}. Only row M=0
// is populated (the 1x64 vector). B(4x16): slot (vgpr v, lane-half h) holds
// row K=2h+v, col N=lane%16. Result row M=0 lands in D VGPR0 of lanes 0-15.
// K=64 -> 16 WMMA steps; N=32 -> 2 tiles -> 32 wmma instructions total.
__global__ void k_out(const float* __restrict__ acc64,
                      const float* __restrict__ W,
                      const float* __restrict__ bias,
                      float* __restrict__ out, float invN) {
    const int lane = threadIdx.x;          // 0..31
    const int half = lane >> 4;            // 0 | 1
    const int nl   = lane & 15;            // N within tile

    v8f c0 = {};
    v8f c1 = {};

    #pragma unroll
    for (int kt = 0; kt < 16; ++kt) {
        const int kbase = kt * 4;
        const int krow  = kbase + 2 * half;   // K rows this lane-half supplies

        // A: only M=0 (nl==0 lanes). half=0 -> K0,K1 ; half=1 -> K2,K3
        v2f a = {0.0f, 0.0f};
        if (nl == 0) {
            a.x = acc64[krow + 0];
            a.y = acc64[krow + 1];
        }

        // B tiles: columns [0,16) and [16,32)
        v2f b0, b1;
        b0.x = W[(krow + 0) * OUT_DIM + nl];
        b0.y = W[(krow + 1) * OUT_DIM + nl];
        b1.x = W[(krow + 0) * OUT_DIM + nl + 16];
        b1.y = W[(krow + 1) * OUT_DIM + nl + 16];

        c0 = __builtin_amdgcn_wmma_f32_16x16x4_f32(
                false, a, false, b0, (short)0, c0, false, false);
        c1 = __builtin_amdgcn_wmma_f32_16x16x4_f32(
                false, a, false, b1, (short)0, c1, false, false);
    }

    // D row M=0 lives in accumulator VGPR0 of lanes 0-15.
    if (half == 0) {
        out[nl]      = c0[0] * invN + bias[nl];
        out[nl + 16] = c1[0] * invN + bias[nl + 16];
    }
}

extern "C" void kernel_launch(void* const* d_in, const int* in_sizes, int n_in,
                              void* d_out, int out_size, void* d_ws, size_t ws_size,
                              hipStream_t stream) {
    const int*   node_seq = (const int*)d_in[0];
    const int*   eidx     = (const int*)d_in[1];   // flat [2, E]
    const float* ew       = (const float*)d_in[2];
    const float* emb      = (const float*)d_in[3]; // [VOCAB, 64]
    const float* W        = (const float*)d_in[4]; // [64, 32]
    const float* bias     = (const float*)d_in[5]; // [32]

    const int N = in_sizes[0];
    const int E = in_sizes[2];
    const int* row = eidx;
    const int* col = eidx + E;

    // workspace layout: deg[N] | dis[N] | s[N] | acc64[64]
    float* deg   = (float*)d_ws;
    float* dis   = deg + N;
    float* s     = dis + N;
    float* acc64 = s + N;

    const int TB = 256;
    const int nb_n = (N + TB - 1) / TB;
    const int nb_e = (E + TB - 1) / TB;

    k_init<<<nb_n, TB, 0, stream>>>(deg, s, acc64, N);
    k_deg <<<nb_e, TB, 0, stream>>>(col, ew, deg, E);
    k_dis <<<nb_n, TB, 0, stream>>>(deg, dis, N);
    k_s   <<<nb_e, TB, 0, stream>>>(row, col, ew, dis, s, E);
    k_nodesum<<<512, TB, 0, stream>>>(node_seq, emb, deg, dis, s, acc64, N);
    k_out <<<1, 32, 0, stream>>>(acc64, W, bias, (float*)d_out, 1.0f / (float)N);
}